// SpatialTransform_bsp_81492709474325
// MI455X (gfx1250) — compile-verified
//
#include <hip/hip_runtime.h>

// ---------------------------------------------------------------------------
// Separable stride-4 B-spline transposed-conv upsampler, (32,2,131,131) f32
// -> (32,2,512,512) f32, implemented as chained V_WMMA_F32_16X16X4_F32 ops.
//
// Per 16x16 tile of each 1-D pass:   D = W(16x8) @ X(8x16), split into two
// K=4 WMMAs. W[t][r] = kern[11-4r+t] when 0<=4r+3-t<=14 else 0 (fixed matrix,
// independent of tile position; W[:,7]==0 so the 8-row window is always
// in-bounds for 16-aligned output blocks).
//
// Memory-bound problem (~103 MB moved, ~170 MFLOP): keep f32, two passes,
// tmp (17.3 MB) stays in L2 (192 MB), final 64 MB stream stored non-temporal.
// ---------------------------------------------------------------------------

typedef float v2f __attribute__((ext_vector_type(2)));
typedef float v8f __attribute__((ext_vector_type(8)));

#define WMMA4(a, b, c) \
  __builtin_amdgcn_wmma_f32_16x16x4_f32(false, (a), false, (b), (short)0, (c), false, false)

#define IN_H   131
#define IN_W   131
#define TW     132            // tmp row pitch (floats): >=131, even for b64 loads
#define OUT_N  512
#define NIMG   64             // 32 * 2 flattened

// W[t][r]: weight of input row (I0/4 + r) for output row (I0 + t)
static __device__ __forceinline__ float bspw(const float* __restrict__ kern, int t, int r) {
  const int s = 4 * r + 3 - t;                  // tap index into flipped kernel
  return (s >= 0 && s <= 14) ? kern[14 - s] : 0.0f;
}

// ---------------- Pass 1: upsample along rows (axis 2) ----------------------
// tiles: img(64) x iblk(32) x wblk(9); one 16x16 tmp tile per wave.
__global__ __launch_bounds__(256) void upsample_rows(
    const float* __restrict__ x, const float* __restrict__ kern,
    float* __restrict__ tmp)
{
  const int lane = threadIdx.x & 31;
  const int tile = blockIdx.x * 8 + (threadIdx.x >> 5);
  const int wb   = tile % 9;
  const int t2   = tile / 9;
  const int ib   = t2 % 32;
  const int img  = t2 / 32;

  const int m    = lane & 15;          // A row (t) / B col / D col
  const int half = lane >> 4;          // selects K-pair {0,1} vs {2,3}
  const int kb   = half * 2;

  // A = W (16x8) split into two 16x4 slabs
  v2f a0, a1;
  a0.x = bspw(kern, m, kb);     a0.y = bspw(kern, m, kb + 1);
  a1.x = bspw(kern, m, kb + 4); a1.y = bspw(kern, m, kb + 5);

  // B = x tile: rows j0..j0+7, 16 columns (zero-fill out of range)
  const int   j0  = ib * 4;
  const int   col = wb * 16 + m;
  const bool  cok = (col < IN_W);
  const float* xi = x + (size_t)img * (IN_H * IN_W);
  auto xv = [&](int row) -> float {
    return (cok && row < IN_H) ? xi[row * IN_W + col] : 0.0f;
  };
  v2f b0, b1;
  b0.x = xv(j0 + kb);     b0.y = xv(j0 + kb + 1);
  b1.x = xv(j0 + kb + 4); b1.y = xv(j0 + kb + 5);

  v8f c = {0.f, 0.f, 0.f, 0.f, 0.f, 0.f, 0.f, 0.f};
  c = WMMA4(a0, b0, c);     // v_wmma_f32_16x16x4_f32
  c = WMMA4(a1, b1, c);

  // D: row I0 + v + 8*half, col = w-tile col; keep cols [0,132) only
  if (col < TW) {
    float* trow = tmp + (size_t)img * (OUT_N * TW)
                      + (size_t)(ib * 16 + half * 8) * TW + col;
#pragma unroll
    for (int v = 0; v < 8; ++v)
      trow[v * TW] = c[v];
  }
}

// ---------------- Pass 2: upsample along columns (axis 3) -------------------
// tiles: img(64) x iblk(32) x jblk(32); one 16x16 output tile per wave.
__global__ __launch_bounds__(256) void upsample_cols(
    const float* __restrict__ tmp, const float* __restrict__ kern,
    float* __restrict__ out)
{
  const int lane = threadIdx.x & 31;
  const int tile = blockIdx.x * 8 + (threadIdx.x >> 5);
  const int jb   = tile & 31;
  const int t2   = tile >> 5;
  const int ib   = t2 & 31;
  const int img  = t2 >> 5;

  const int m    = lane & 15;          // A row (output I) / B col (phase t)
  const int half = lane >> 4;
  const int kb   = half * 2;

  // A = tmp tile: 16 rows (I0..I0+15) x 8 cols (j0..j0+7); b64 paired loads
  const float* arow = tmp + (size_t)img * (OUT_N * TW)
                          + (size_t)(ib * 16 + m) * TW + jb * 4;
  v2f a0 = *(const v2f*)(arow + kb);
  v2f a1 = *(const v2f*)(arow + kb + 4);

  // B = W^T (8x16): B[k][n] = W[n][k]; per-lane identical scalars to pass-1 A
  v2f b0, b1;
  b0.x = bspw(kern, m, kb);     b0.y = bspw(kern, m, kb + 1);
  b1.x = bspw(kern, m, kb + 4); b1.y = bspw(kern, m, kb + 5);

  v8f c = {0.f, 0.f, 0.f, 0.f, 0.f, 0.f, 0.f, 0.f};
  c = WMMA4(a0, b0, c);
  c = WMMA4(a1, b1, c);

  // 64 MB streaming output: non-temporal stores so tmp stays L2-resident
  float* orow = out + (size_t)img * (OUT_N * OUT_N)
                    + (size_t)(ib * 16 + half * 8) * OUT_N + jb * 16 + m;
#pragma unroll
  for (int v = 0; v < 8; ++v)
    __builtin_nontemporal_store(c[v], orow + v * OUT_N);
}

extern "C" void kernel_launch(void* const* d_in, const int* in_sizes, int n_in,
                              void* d_out, int out_size, void* d_ws, size_t ws_size,
                              hipStream_t stream) {
  const float* x    = (const float*)d_in[0];   // (32,2,131,131) f32
  const float* kern = (const float*)d_in[1];   // (15,) f32
  float* out = (float*)d_out;                  // (32,2,512,512) f32
  float* tmp = (float*)d_ws;                   // needs 64*512*132*4 = 17,301,504 B

  // pass 1: 64*32*9 = 18432 tiles, 8 waves/block -> 2304 blocks
  upsample_rows<<<dim3(2304), dim3(256), 0, stream>>>(x, kern, tmp);
  // pass 2: 64*32*32 = 65536 tiles -> 8192 blocks
  upsample_cols<<<dim3(8192), dim3(256), 0, stream>>>(tmp, kern, out);
}